// GCN_83502754169570
// MI455X (gfx1250) — compile-verified
//
#include <hip/hip_runtime.h>
#include <hip/hip_bf16.h>
#include <math.h>

typedef __attribute__((ext_vector_type(2))) float v2f;
typedef __attribute__((ext_vector_type(8))) float v8f;

#define NEG_SLOPE 0.2f

// ---------------------------------------------------------------------------
// Degree / normalization kernels
// ---------------------------------------------------------------------------
__global__ void k_init_deg(float* __restrict__ deg, int n) {
  int i = blockIdx.x * blockDim.x + threadIdx.x;
  if (i < n) deg[i] = 1.0f;  // self-loop weight 1
}

__global__ void k_accum_deg(const int* __restrict__ dst, const float* __restrict__ w,
                            float* __restrict__ deg, int E) {
  int e = blockIdx.x * blockDim.x + threadIdx.x;
  if (e < E) atomicAdd(&deg[dst[e]], w[e]);
}

__global__ void k_deg_to_dinv(float* __restrict__ deg, int n) {
  int i = blockIdx.x * blockDim.x + threadIdx.x;
  if (i < n) {
    float d = deg[i];
    deg[i] = (d > 0.0f) ? rsqrtf(d) : 0.0f;
  }
}

// ---------------------------------------------------------------------------
// FP32 WMMA GEMM: C[M,N] = A[M,K] @ B[K,N]
// Block = 128 threads = 4 waves; each wave -> one 16-row tile x 64 cols.
// grid.x covers rows in chunks of 64, grid.y covers cols in chunks of 64.
// Uses V_WMMA_F32_16X16X4_F32 (wave32, fp32 in/out).
// A-fragment layout (ISA 7.12.2): lanes 0-15 hold M=0..15; VGPR0 = K {0,2},
// VGPR1 = K {1,3} selected by lane half. B/C: rows striped across lane halves.
// ---------------------------------------------------------------------------
__global__ void k_gemm_wmma_f32(const float* __restrict__ A, const float* __restrict__ B,
                                float* __restrict__ C, int M, int K, int N) {
  const int lane = threadIdx.x & 31;
  const int wave = threadIdx.x >> 5;
  const int row0 = blockIdx.x * 64 + wave * 16;
  if (row0 >= M) return;                       // wave-uniform: EXEC stays all-ones
  const int col0 = blockIdx.y * 64;
  const int half = lane >> 4;                  // 0: lanes 0-15, 1: lanes 16-31
  const int l    = lane & 15;
  const int arow = row0 + l;

  v8f c0 = {}, c1 = {}, c2 = {}, c3 = {};

  for (int k0 = 0; k0 < K; k0 += 4) {
    const int kk = k0 + (half << 1);           // K offset for this lane half
    v2f a = *(const v2f*)(A + (size_t)arow * K + kk);

    const float* Bp = B + (size_t)kk * N + col0 + l;
    v2f b0 = { Bp[0],  Bp[N + 0]  };
    v2f b1 = { Bp[16], Bp[N + 16] };
    v2f b2 = { Bp[32], Bp[N + 32] };
    v2f b3 = { Bp[48], Bp[N + 48] };

    c0 = __builtin_amdgcn_wmma_f32_16x16x4_f32(false, a, false, b0, (short)0, c0, false, false);
    c1 = __builtin_amdgcn_wmma_f32_16x16x4_f32(false, a, false, b1, (short)0, c1, false, false);
    c2 = __builtin_amdgcn_wmma_f32_16x16x4_f32(false, a, false, b2, (short)0, c2, false, false);
    c3 = __builtin_amdgcn_wmma_f32_16x16x4_f32(false, a, false, b3, (short)0, c3, false, false);
  }

#pragma unroll
  for (int r = 0; r < 8; ++r) {
    const int row = row0 + r + half * 8;       // C VGPR r: M=r (lanes 0-15) / M=r+8
    float* Cp = C + (size_t)row * N + col0 + l;
    Cp[0]  = c0[r];
    Cp[16] = c1[r];
    Cp[32] = c2[r];
    Cp[48] = c3[r];
  }
}

// ---------------------------------------------------------------------------
// Self-loop init: out[i,:] = xw[i,:] * dinv[i]^2   (norm of self edge, w=1)
// ---------------------------------------------------------------------------
template <int F>
__global__ void k_selfloop_init(const float* __restrict__ xw, const float* __restrict__ dinv,
                                float* __restrict__ out, int n) {
  size_t idx = (size_t)blockIdx.x * blockDim.x + threadIdx.x;
  size_t total = (size_t)n * F;
  if (idx < total) {
    int node = (int)(idx / F);
    float di = dinv[node];
    out[idx] = xw[idx] * di * di;
  }
}

// ---------------------------------------------------------------------------
// Edge scatter: one wave per edge; lanes cover F features with float4/float2.
// out[dst,:] += xw[src,:] * (dinv[src]*w*dinv[dst]).  Atomic bursts are
// 32 consecutive lanes -> contiguous cachelines, serviced in L2 (xw fits L2).
// ---------------------------------------------------------------------------
template <int F>
__global__ void k_scatter(const int* __restrict__ src, const int* __restrict__ dst,
                          const float* __restrict__ w, const float* __restrict__ dinv,
                          const float* __restrict__ xw, float* __restrict__ out, int E) {
  int e = blockIdx.x * (blockDim.x >> 5) + (threadIdx.x >> 5);
  if (e >= E) return;
  const int lane = threadIdx.x & 31;
  const int s = src[e];
  const int d = dst[e];
  const float norm = dinv[s] * w[e] * dinv[d];
  const float* xp = xw + (size_t)s * F;
  float* op = out + (size_t)d * F;
  if (F == 128) {
    float4 v = *(const float4*)(xp + lane * 4);
    atomicAdd(op + lane * 4 + 0, v.x * norm);
    atomicAdd(op + lane * 4 + 1, v.y * norm);
    atomicAdd(op + lane * 4 + 2, v.z * norm);
    atomicAdd(op + lane * 4 + 3, v.w * norm);
  } else {  // F == 64
    float2 v = *(const float2*)(xp + lane * 2);
    atomicAdd(op + lane * 2 + 0, v.x * norm);
    atomicAdd(op + lane * 2 + 1, v.y * norm);
  }
}

// ---------------------------------------------------------------------------
// Bias + LeakyReLU, in place.  F must be a power of two.
// ---------------------------------------------------------------------------
template <int F>
__global__ void k_bias_lrelu(float* __restrict__ h, const float* __restrict__ b, int n) {
  size_t idx = (size_t)blockIdx.x * blockDim.x + threadIdx.x;
  size_t total = (size_t)n * F;
  if (idx < total) {
    float v = h[idx] + b[idx & (F - 1)];
    h[idx] = (v > 0.0f) ? v : NEG_SLOPE * v;
  }
}

// ---------------------------------------------------------------------------
// Row-wise log_softmax over 64 classes (+ bias b2): one wave per row,
// 2 values per lane, wave32 shuffle reductions.
// ---------------------------------------------------------------------------
__global__ void k_logsoftmax64(const float* __restrict__ in, const float* __restrict__ b2,
                               float* __restrict__ out, int n) {
  int row = blockIdx.x * (blockDim.x >> 5) + (threadIdx.x >> 5);
  if (row >= n) return;
  const int lane = threadIdx.x & 31;
  const float* ip = in + (size_t)row * 64;
  float v0 = ip[lane]      + b2[lane];
  float v1 = ip[lane + 32] + b2[lane + 32];

  float m = fmaxf(v0, v1);
#pragma unroll
  for (int off = 16; off > 0; off >>= 1) m = fmaxf(m, __shfl_xor(m, off, 32));

  float s = expf(v0 - m) + expf(v1 - m);
#pragma unroll
  for (int off = 16; off > 0; off >>= 1) s += __shfl_xor(s, off, 32);

  const float lse = m + logf(s);
  out[(size_t)row * 64 + lane]      = v0 - lse;
  out[(size_t)row * 64 + lane + 32] = v1 - lse;
}

// ---------------------------------------------------------------------------
static inline int ceil_div(int a, int b) { return (a + b - 1) / b; }

extern "C" void kernel_launch(void* const* d_in, const int* in_sizes, int n_in,
                              void* d_out, int out_size, void* d_ws, size_t ws_size,
                              hipStream_t stream) {
  const float* x  = (const float*)d_in[0];
  const int*   ei = (const int*)d_in[1];     // [2, E] flat: src then dst
  const float* w  = (const float*)d_in[2];
  const float* W1 = (const float*)d_in[3];
  const float* b1 = (const float*)d_in[4];
  const float* W2 = (const float*)d_in[5];
  const float* b2 = (const float*)d_in[6];

  const int F1 = 128, F2 = 64;
  const int E = in_sizes[2];
  const int N = in_sizes[0] / F1;            // 100000
  const int* src = ei;
  const int* dst = ei + E;

  // Workspace carve-out (256B-aligned slices).
  char* ws = (char*)d_ws;
  size_t off = 0;
  auto take = [&](size_t bytes) -> float* {
    float* p = (float*)(ws + off);
    off += (bytes + 255) & ~(size_t)255;
    return p;
  };
  float* dinv = take((size_t)N * sizeof(float));
  float* xw1  = take((size_t)N * F1 * sizeof(float));
  float* h1   = take((size_t)N * F1 * sizeof(float));   // aggregated layer 1 -> activated
  float* xw2  = take((size_t)N * F2 * sizeof(float));
  float* agg2 = take((size_t)N * F2 * sizeof(float));
  (void)ws_size;

  const int T = 256;

  // 1) degrees (self-loop = 1) -> dinv
  k_init_deg<<<ceil_div(N, T), T, 0, stream>>>(dinv, N);
  k_accum_deg<<<ceil_div(E, T), T, 0, stream>>>(dst, w, dinv, E);
  k_deg_to_dinv<<<ceil_div(N, T), T, 0, stream>>>(dinv, N);

  // 2) xw1 = x @ W1   (M=100000, K=128, N=128)
  {
    dim3 grid(ceil_div(N, 64), F1 / 64);
    k_gemm_wmma_f32<<<grid, 128, 0, stream>>>(x, W1, xw1, N, F1, F1);
  }

  // 3) layer-1 aggregation: self loops then edge scatter
  k_selfloop_init<128><<<ceil_div(N * F1, T), T, 0, stream>>>(xw1, dinv, h1, N);
  k_scatter<128><<<ceil_div(E, T / 32), T, 0, stream>>>(src, dst, w, dinv, xw1, h1, E);

  // 4) bias + leaky relu (in place)
  k_bias_lrelu<128><<<ceil_div(N * F1, T), T, 0, stream>>>(h1, b1, N);

  // 5) xw2 = h1 @ W2  (M=100000, K=128, N=64)
  {
    dim3 grid(ceil_div(N, 64), F2 / 64);
    k_gemm_wmma_f32<<<grid, 128, 0, stream>>>(h1, W2, xw2, N, F1, F2);
  }

  // 6) layer-2 aggregation
  k_selfloop_init<64><<<ceil_div(N * F2, T), T, 0, stream>>>(xw2, dinv, agg2, N);
  k_scatter<64><<<ceil_div(E, T / 32), T, 0, stream>>>(src, dst, w, dinv, xw2, agg2, E);

  // 7) bias + log_softmax -> d_out
  k_logsoftmax64<<<ceil_div(N, T / 32), T, 0, stream>>>(agg2, b2, (float*)d_out, N);
}